// TriplaneDeformableAttention_721554505956
// MI455X (gfx1250) — compile-verified
//
#include <hip/hip_runtime.h>
#include <math.h>

#define N_Q   100000
#define C_CH  128
#define HH    256
#define WW    256
#define POSD  36
#define PLANE_ELEMS (HH * WW * C_CH)   // 8,388,608 floats per plane

typedef __attribute__((ext_vector_type(2))) float v2f;
typedef __attribute__((ext_vector_type(4))) float f4;
typedef __attribute__((ext_vector_type(4))) int   i4;
typedef __attribute__((ext_vector_type(8))) float v8f;

// 32-byte per-tap record, fully precomputed in phase 1.
struct __align__(32) TapRec {
    int   off;   // absolute element offset of (y0,x0) tap in featT (plane base folded in)
    int   dx;    // step to x1 neighbor: 0 or C_CH
    int   dy;    // step to y1 neighbor: 0 or WW*C_CH
    int   pad;
    float w00, w01, w10, w11;  // bilinear weights * attention weight
};

struct P1Params {
    const float* query;
    const float* pe;
    const float* Wa[3];
    const float* ba[3];
    const float* Wo[3];
    const float* bo[3];
    TapRec*      taps;
};

// ---------------------------------------------------------------------------
// Kernel 0: transpose one plane [C=128][HW=65536] -> [HW][C] (channel-last)
// ---------------------------------------------------------------------------
__global__ __launch_bounds__(256) void tr_kernel(const float* __restrict__ in,
                                                 float* __restrict__ outp) {
    __shared__ float tile[32][33];
    const int tx  = threadIdx.x & 31;
    const int ty  = threadIdx.x >> 5;          // 0..7
    const int hw0 = blockIdx.x * 32;           // 2048 blocks
    const int c0  = blockIdx.y * 32;           // 4 blocks
#pragma unroll
    for (int i = 0; i < 4; ++i) {
        int c = c0 + ty + i * 8;
        tile[ty + i * 8][tx] = in[(size_t)c * (HH * WW) + hw0 + tx];   // coalesced read
    }
    __syncthreads();
#pragma unroll
    for (int i = 0; i < 4; ++i) {
        int hw = hw0 + ty + i * 8;
        outp[(size_t)hw * C_CH + c0 + tx] = tile[tx][ty + i * 8];      // coalesced write
    }
}

// ---------------------------------------------------------------------------
// Kernel 1: WMMA projection pe[16,36] x [Wa|Wo][36,12], softmax over K=4,
// tap-record build. One wave per (16-query tile, plane).
// ---------------------------------------------------------------------------
// Branch-free B-fragment fetch: both loads are always in-bounds
// (Wa: 36*4=144 elems, idx <= 35*4+3 = 143; Wo: 36*8=288 elems, idx <= 287),
// selection via v_cndmask instead of divergent exec-mask branches.
__device__ __forceinline__ float wcol(const float* Wa, const float* Wo, int k, int nn) {
    float va = Wa[k * 4 + (nn & 3)];
    float vo = Wo[k * 8 + ((nn + 4) & 7)];   // (nn-4) mod 8 for nn in 4..11
    float r  = (nn < 4) ? va : vo;
    return (nn < 12) ? r : 0.0f;             // cols 12..15 are zero padding
}

__global__ __launch_bounds__(32) void p1_kernel(P1Params prm) {
    const int p    = blockIdx.y;           // plane 0..2
    const int tile = blockIdx.x;           // 16 queries per tile, N_Q/16 tiles
    const int lane = threadIdx.x;          // wave32
    const int m    = lane & 15;
    const int kb   = (lane >> 4) << 1;     // 0 for lanes 0-15, 2 for lanes 16-31
    const float* Wa    = prm.Wa[p];
    const float* Wo    = prm.Wo[p];
    const float* peRow = prm.pe + (size_t)(tile * 16 + m) * POSD;

    v8f acc = {0.f, 0.f, 0.f, 0.f, 0.f, 0.f, 0.f, 0.f};
#pragma unroll
    for (int kc = 0; kc < 9; ++kc) {       // K = 36 = 9 chunks of 4
        const int k0 = kc * 4 + kb;
        v2f a;                              // A 16x4 f32 fragment (ISA layout)
        a.x = peRow[k0];
        a.y = peRow[k0 + 1];
        v2f b;                              // B 4x16 f32 fragment
        b.x = wcol(Wa, Wo, k0,     m);
        b.y = wcol(Wa, Wo, k0 + 1, m);
        acc = __builtin_amdgcn_wmma_f32_16x16x4_f32(
            /*neg_a=*/false, a, /*neg_b=*/false, b,
            /*c_mod=*/(short)0, acc, /*reuse_a=*/false, /*reuse_b=*/false);
    }

    __shared__ float S[16][16];
    const int rbase = (lane >> 4) * 8;
#pragma unroll
    for (int r = 0; r < 8; ++r) S[rbase + r][m] = acc[r];
    __syncthreads();

    if (lane < 16) {
        const int q = tile * 16 + lane;
        const float* ba = prm.ba[p];
        const float* bo = prm.bo[p];
        // softmax over 4 attention logits
        float av[4];
        float mx = -3.4e38f;
#pragma unroll
        for (int k = 0; k < 4; ++k) { av[k] = S[lane][k] + ba[k]; mx = fmaxf(mx, av[k]); }
        float ssum = 0.f;
#pragma unroll
        for (int k = 0; k < 4; ++k) { av[k] = __expf(av[k] - mx); ssum += av[k]; }
        const float inv = 1.0f / ssum;

        // plane dim selection: xy->(0,1), xz->(0,2), yz->(1,2); gy=dim0, gx=dim1
        const int d0 = (p == 2) ? 1 : 0;
        const int d1 = (p == 0) ? 1 : 2;
        const float qy = prm.query[q * 3 + d0];
        const float qx = prm.query[q * 3 + d1];
        const int pbase = p * PLANE_ELEMS;
        TapRec* tp = prm.taps + (size_t)q * 12 + p * 4;
#pragma unroll
        for (int k = 0; k < 4; ++k) {
            const float gy = qy + S[lane][4 + 2 * k]     + bo[2 * k];
            const float gx = qx + S[lane][4 + 2 * k + 1] + bo[2 * k + 1];
            // grid_sample align_corners=False, border padding
            float x = fminf(fmaxf(((gx + 1.0f) * (float)WW - 1.0f) * 0.5f, 0.0f), (float)(WW - 1));
            float y = fminf(fmaxf(((gy + 1.0f) * (float)HH - 1.0f) * 0.5f, 0.0f), (float)(HH - 1));
            float x0f = floorf(x), y0f = floorf(y);
            float wx = x - x0f,   wy = y - y0f;
            int xi0 = (int)x0f,   yi0 = (int)y0f;
            int dx = (xi0 < WW - 1) ? C_CH : 0;
            int dy = (yi0 < HH - 1) ? WW * C_CH : 0;
            float a = av[k] * inv;
            i4 ri; ri.x = pbase + (yi0 * WW + xi0) * C_CH; ri.y = dx; ri.z = dy; ri.w = 0;
            f4 rw;
            rw.x = (1.f - wx) * (1.f - wy) * a;
            rw.y = wx * (1.f - wy) * a;
            rw.z = (1.f - wx) * wy * a;
            rw.w = wx * wy * a;
            *(i4*)(&tp[k]) = ri;
            *(f4*)((char*)&tp[k] + 16) = rw;
        }
    }
}

// ---------------------------------------------------------------------------
// Kernel 2: gather + weighted accumulate. One wave per query, lane l handles
// channels 4l..4l+3 as float4. All neighbor loads are 512B wave-contiguous.
// ---------------------------------------------------------------------------
__global__ __launch_bounds__(256) void p2_kernel(const TapRec* __restrict__ taps,
                                                 const float* __restrict__ featT,
                                                 float* __restrict__ out) {
    const int lane = threadIdx.x & 31;
    const int wv   = threadIdx.x >> 5;     // 8 waves/block
    const int n    = blockIdx.x * 8 + wv;  // N_Q/8 blocks, exact
    const TapRec* tp = taps + (size_t)n * 12;

    f4 acc = {0.f, 0.f, 0.f, 0.f};
#pragma unroll
    for (int t = 0; t < 12; ++t) {
        i4 ri = *(const i4*)(&tp[t]);                          // broadcast loads
        f4 rw = *(const f4*)((const char*)&tp[t] + 16);
        const float* b = featT + ri.x + lane * 4;
        f4 v00 = *(const f4*)(b);
        f4 v01 = *(const f4*)(b + ri.y);
        f4 v10 = *(const f4*)(b + ri.z);
        f4 v11 = *(const f4*)(b + ri.y + ri.z);
        acc += rw.x * v00 + rw.y * v01 + rw.z * v10 + rw.w * v11;
    }
    *(f4*)(out + (size_t)n * C_CH + lane * 4) = acc;
}

// ---------------------------------------------------------------------------
extern "C" void kernel_launch(void* const* d_in, const int* in_sizes, int n_in,
                              void* d_out, int out_size, void* d_ws, size_t ws_size,
                              hipStream_t stream) {
    // setup_inputs() dict order:
    // 0:query 1:pe | per plane p: 2+5p:feat 3+5p:Wa 4+5p:ba 5+5p:Wo 6+5p:bo
    const float* feat[3] = {(const float*)d_in[2], (const float*)d_in[7], (const float*)d_in[12]};
    P1Params prm;
    prm.query = (const float*)d_in[0];
    prm.pe    = (const float*)d_in[1];
    for (int p = 0; p < 3; ++p) {
        prm.Wa[p] = (const float*)d_in[3 + p * 5];
        prm.ba[p] = (const float*)d_in[4 + p * 5];
        prm.Wo[p] = (const float*)d_in[5 + p * 5];
        prm.bo[p] = (const float*)d_in[6 + p * 5];
    }

    // Workspace layout: [featT: 3*PLANE_ELEMS floats = 96MB][taps: N_Q*12*32B = 38.4MB]
    float*  featT = (float*)d_ws;
    TapRec* taps  = (TapRec*)((char*)d_ws + (size_t)3 * PLANE_ELEMS * sizeof(float));
    prm.taps = taps;

    for (int p = 0; p < 3; ++p)
        tr_kernel<<<dim3(2048, 4), 256, 0, stream>>>(feat[p], featT + (size_t)p * PLANE_ELEMS);

    p1_kernel<<<dim3(N_Q / 16, 3), 32, 0, stream>>>(prm);

    p2_kernel<<<N_Q / 8, 256, 0, stream>>>(taps, featT, (float*)d_out);
}